// RenG_10986526343472
// MI455X (gfx1250) — compile-verified
//
#include <hip/hip_runtime.h>
#include <math.h>
#include <stdint.h>

// ---------------- problem constants ----------------
#define B_   65536
#define N_   256
#define L_   256
#define M_   96
#define P_   128
#define EPS_C 0.001f
// GAMMA == 1.0 exploited analytically: Q=-I, R=I, S=0, Lq=Lr=I.

// ---------------- CDNA5 WMMA types ----------------
typedef float v2f __attribute__((ext_vector_type(2)));
typedef float v8f __attribute__((ext_vector_type(8)));

// D = A(16x4 f32) x B(4x16 f32) + C, fp32 WMMA (wave32).
#define WMMA4(ACC, APTR, BPTR)                                                   \
  do {                                                                           \
    v2f _a = *(const v2f*)(APTR);                                                \
    v2f _b = *(const v2f*)(BPTR);                                                \
    ACC = __builtin_amdgcn_wmma_f32_16x16x4_f32(false, _a, false, _b,            \
                                                (short)0, ACC, false, false);    \
  } while (0)

// One A fragment feeds two N-tiles (A reuse, back-to-back independent WMMAs).
#define WMMA4_PAIR(ACC0, ACC1, APTR, B0PTR, B1PTR)                               \
  do {                                                                           \
    v2f _a  = *(const v2f*)(APTR);                                               \
    v2f _b0 = *(const v2f*)(B0PTR);                                              \
    v2f _b1 = *(const v2f*)(B1PTR);                                              \
    ACC0 = __builtin_amdgcn_wmma_f32_16x16x4_f32(false, _a, false, _b0,          \
                                                 (short)0, ACC0, false, false);  \
    ACC1 = __builtin_amdgcn_wmma_f32_16x16x4_f32(false, _a, false, _b1,          \
                                                 (short)0, ACC1, false, false);  \
  } while (0)

// Async memory -> LDS copy (16B per lane), tracked by ASYNCcnt (CDNA5 §08).
__device__ __forceinline__ void async_copy16(uint32_t lds_off, const float* g) {
  asm volatile("global_load_async_to_lds_b128 %0, %1, off"
               :: "v"(lds_off), "v"(g) : "memory");
}
__device__ __forceinline__ void async_wait0() {
  asm volatile("s_wait_asynccnt 0x0" ::: "memory");
}

// =====================================================================
// Phase A kernels: small linear algebra (set_param)
// =====================================================================

// IpM = I + Mm = X3^T X3 + Y3 - Y3^T + Z3^T Z3 + (1+eps) I   (96x96)
__global__ void k_build_IpM(const float* __restrict__ X3, const float* __restrict__ Y3,
                            const float* __restrict__ Z3, float* __restrict__ IpM) {
  int idx = blockIdx.x * blockDim.x + threadIdx.x;
  if (idx >= 96 * 96) return;
  int i = idx / 96, j = idx % 96;
  float s = 0.f;
  for (int k = 0; k < 96; ++k) s += X3[k * 96 + i] * X3[k * 96 + j];
  for (int k = 0; k < 32; ++k) s += Z3[k * 96 + i] * Z3[k * 96 + j];
  s += Y3[i * 96 + j] - Y3[j * 96 + i];
  if (i == j) s += 1.0f + EPS_C;
  IpM[idx] = s;
}

// Generic small GEMM: C = alpha*op(A)op(B) [+ beta*C] [+ idalpha*I]
__global__ void k_gemm(float* __restrict__ C, const float* __restrict__ A,
                       const float* __restrict__ Bm, int M, int N, int K,
                       int lda, int ldb, int ldc, int tA, int tB,
                       float alpha, float beta, float idalpha) {
  int idx = blockIdx.x * blockDim.x + threadIdx.x;
  if (idx >= M * N) return;
  int i = idx / N, j = idx % N;
  float s = 0.f;
  for (int k = 0; k < K; ++k) {
    float a = tA ? A[k * lda + i] : A[i * lda + k];
    float b = tB ? Bm[j * ldb + k] : Bm[k * ldb + j];
    s += a * b;
  }
  float c = alpha * s;
  if (beta != 0.f) c += beta * C[i * ldc + j];
  if (idalpha != 0.f && i == j) c += idalpha;
  C[i * ldc + j] = c;
}

// C = alpha*A [+ beta*C]   (block copy with strides)
__global__ void k_scale_add(float* __restrict__ C, const float* __restrict__ A,
                            int ldc, int lda, int M, int N, float alpha, float beta) {
  int idx = blockIdx.x * blockDim.x + threadIdx.x;
  if (idx >= M * N) return;
  int i = idx / N, j = idx % N;
  float v = alpha * A[i * lda + j];
  if (beta != 0.f) v += beta * C[i * ldc + j];
  C[i * ldc + j] = v;
}

__global__ void k_fill(float* __restrict__ C, int ldc, int M, int N, float v) {
  int idx = blockIdx.x * blockDim.x + threadIdx.x;
  if (idx >= M * N) return;
  C[(idx / N) * ldc + (idx % N)] = v;
}

// One-block Gauss-Jordan inverse (no pivoting), n <= 256, aug is n x 2n scratch.
__global__ __launch_bounds__(1024) void k_gj_inverse(const float* __restrict__ A,
                                                     float* __restrict__ Ainv,
                                                     float* __restrict__ aug, int n) {
  const int tid = threadIdx.x;
  const int n2 = 2 * n;
  __shared__ float prow[512];
  __shared__ float pivinv_s;
  for (int idx = tid; idx < n * n2; idx += blockDim.x) {
    int i = idx / n2, j = idx % n2;
    aug[idx] = (j < n) ? A[i * n + j] : ((j - n == i) ? 1.f : 0.f);
  }
  __syncthreads();
  for (int p = 0; p < n; ++p) {
    if (tid == 0) pivinv_s = 1.f / aug[p * n2 + p];
    __syncthreads();
    float piv = pivinv_s;
    for (int j = tid; j < n2; j += blockDim.x) {
      float v = aug[p * n2 + j] * piv;
      aug[p * n2 + j] = v;
      prow[j] = v;
    }
    __syncthreads();
    int row = tid >> 2;
    int jc = tid & 3;
    float f = 0.f;
    if (row < n && row != p) f = aug[row * n2 + p];
    __syncthreads();
    if (row < n && row != p) {
      int q = n2 >> 2;
      int j0 = jc * q;
      for (int j = j0; j < j0 + q; ++j) aug[row * n2 + j] -= f * prow[j];
    }
    __syncthreads();
  }
  for (int idx = tid; idx < n * n; idx += blockDim.x)
    Ainv[idx] = aug[(idx / n) * n2 + n + (idx % n)];
}

// E = 0.5*(H11 + H33 + Y - Y^T)   (256x256, H is 768x768)
__global__ void k_build_E(const float* __restrict__ H, const float* __restrict__ Y,
                          float* __restrict__ E) {
  int idx = blockIdx.x * blockDim.x + threadIdx.x;
  if (idx >= 256 * 256) return;
  int i = idx / 256, j = idx % 256;
  E[idx] = 0.5f * (H[i * 768 + j] + H[(512 + i) * 768 + 512 + j] +
                   Y[i * 256 + j] - Y[j * 256 + i]);
}

// D11 = -tril(H22,-1), Lam = 0.5*diag(H22)
__global__ void k_build_D11Lam(const float* __restrict__ H, float* __restrict__ D11,
                               float* __restrict__ Lam) {
  int idx = blockIdx.x * blockDim.x + threadIdx.x;
  if (idx >= 256 * 256) return;
  int i = idx / 256, j = idx % 256;
  float h = H[(256 + i) * 768 + 256 + j];
  D11[idx] = (j < i) ? -h : 0.f;
  if (i == j) Lam[i] = 0.5f * h;
}

// =====================================================================
// Phase B: batched forward with fp32 WMMA (the big kernel)
// =====================================================================
#define TB 64
#define LDSX 260   // padded stride for 256-wide tiles (260 % 64 == 4 -> conflict-free)
#define LDSU 100   // padded stride for 96-wide tile

__global__ __launch_bounds__(256)
void k_forward(const float* __restrict__ x, const float* __restrict__ u,
               const float* __restrict__ Wb,   // 256 x 352 = [C1 | D12]
               const float* __restrict__ Wout, // 384 x 608 = [[F2|B1_2|B2_2];[C2|0|D22]]
               const float* __restrict__ D11,  // 256 x 256 strictly lower
               const float* __restrict__ Lam,  // 256
               float* __restrict__ outY, float* __restrict__ outX) {
  __shared__ float sX[TB * LDSX];
  __shared__ float sE[TB * LDSX];  // holds base, then epsilon
  __shared__ float sU[TB * LDSU];

  const int tid = threadIdx.x;
  const int wave = tid >> 5;
  const int lane = tid & 31;
  const int half = lane >> 4;  // 0/1
  const int lrow = lane & 15;
  const int b0 = blockIdx.x * TB;

  // warm L2 with weight panels (global_prefetch_b8)
  __builtin_prefetch(Wout + (size_t)tid * 64, 0, 2);
  __builtin_prefetch(Wb + (size_t)tid * 32, 0, 2);

  // ---- async tile fills: x (64x256), u (64x96) -> LDS (ASYNCcnt path) ----
  {
    int r = tid >> 2;
    int cx = (tid & 3) * 64;
    const float* gx = x + (size_t)(b0 + r) * 256 + cx;
    uint32_t lx = (uint32_t)(size_t)&sX[r * LDSX + cx];
#pragma unroll
    for (int q = 0; q < 16; ++q) async_copy16(lx + q * 16, gx + q * 4);
    int cu = (tid & 3) * 24;
    const float* gu = u + (size_t)(b0 + r) * 96 + cu;
    uint32_t lu = (uint32_t)(size_t)&sU[r * LDSU + cu];
#pragma unroll
    for (int q = 0; q < 6; ++q) async_copy16(lu + q * 16, gu + q * 4);
  }
  async_wait0();
  __syncthreads();

  // ---- Stage 1: base = [x|u] @ Wb^T -> sE ; dual N-tile per A fragment ----
  for (int t = wave; t < 32; t += 8) {
    int m0 = (t >> 3) * 16;
    int n0 = (t & 7) * 32;
    v8f acc0 = {}, acc1 = {};
    const float* wb0 = Wb + (size_t)(n0 + lrow) * 352 + 2 * half;
    const float* wb1 = wb0 + 16 * 352;
    const float* ax = &sX[(m0 + lrow) * LDSX + 2 * half];
    const float* au = &sU[(m0 + lrow) * LDSU + 2 * half];
#pragma unroll 4
    for (int k = 0; k < 256; k += 4) WMMA4_PAIR(acc0, acc1, ax + k, wb0 + k, wb1 + k);
#pragma unroll 4
    for (int k = 0; k < 96; k += 4)
      WMMA4_PAIR(acc0, acc1, au + k, wb0 + 256 + k, wb1 + 256 + k);
#pragma unroll
    for (int r = 0; r < 8; ++r) {
      sE[(m0 + r + 8 * half) * LDSX + n0 + lrow] = acc0[r];
      sE[(m0 + r + 8 * half) * LDSX + n0 + 16 + lrow] = acc1[r];
    }
  }
  __syncthreads();

  // ---- Stage 2: blocked triangular tanh scan; waves 0..3 own one M-tile each ----
  if (wave < 4) {
    const int m0 = wave * 16;
    for (int blk = 0; blk < 16; ++blk) {
      const int i0 = blk * 16;
      v8f acc;
#pragma unroll
      for (int r = 0; r < 8; ++r)
        acc[r] = sE[(m0 + r + 8 * half) * LDSX + i0 + lrow];  // base cols
      // cross-block: eps[:, :i0] @ D11[i0+n, :i0]^T   (WMMA)
      const float* dw = D11 + (size_t)(i0 + lrow) * 256 + 2 * half;
      const float* ae = &sE[(m0 + lrow) * LDSX + 2 * half];
#pragma unroll 4
      for (int k = 0; k < i0; k += 4) WMMA4(acc, ae + k, dw + k);
      // diagonal 16x16 block of D11 into registers (lane holds row i0+lrow)
      float d11r[16];
      {
        const float4* dp = (const float4*)(D11 + (size_t)(i0 + lrow) * 256 + i0);
#pragma unroll
        for (int q = 0; q < 4; ++q) {
          float4 v = dp[q];
          d11r[q * 4 + 0] = v.x; d11r[q * 4 + 1] = v.y;
          d11r[q * 4 + 2] = v.z; d11r[q * 4 + 3] = v.w;
        }
      }
      const float lamv = Lam[i0 + lrow];
      // serial 16 steps within the block
#pragma unroll
      for (int j = 0; j < 16; ++j) {
        if (lrow == j) {
#pragma unroll
          for (int r = 0; r < 8; ++r) {
            float e = tanhf(acc[r] / lamv);
            sE[(m0 + r + 8 * half) * LDSX + i0 + j] = e;
          }
        }
        if (lrow > j) {
#pragma unroll
          for (int r = 0; r < 8; ++r) {
            float e = sE[(m0 + r + 8 * half) * LDSX + i0 + j];
            acc[r] += e * d11r[j];
          }
        }
      }
    }
  }
  __syncthreads();

  // ---- Stage 3: out = [x|eps|u] @ Wout^T ; dual N-tile per A fragment ----
  for (int t = wave; t < 48; t += 8) {
    int m0 = (t / 12) * 16;
    int n0 = (t % 12) * 32;   // pair covers n0 and n0+16; never straddles col 256
    v8f acc0 = {}, acc1 = {};
    const float* wo0 = Wout + (size_t)(n0 + lrow) * 608 + 2 * half;
    const float* wo1 = wo0 + 16 * 608;
    const float* ax = &sX[(m0 + lrow) * LDSX + 2 * half];
    const float* ae = &sE[(m0 + lrow) * LDSX + 2 * half];
    const float* au = &sU[(m0 + lrow) * LDSU + 2 * half];
#pragma unroll 4
    for (int k = 0; k < 256; k += 4) WMMA4_PAIR(acc0, acc1, ax + k, wo0 + k, wo1 + k);
#pragma unroll 4
    for (int k = 0; k < 256; k += 4)
      WMMA4_PAIR(acc0, acc1, ae + k, wo0 + 256 + k, wo1 + 256 + k);
#pragma unroll 4
    for (int k = 0; k < 96; k += 4)
      WMMA4_PAIR(acc0, acc1, au + k, wo0 + 512 + k, wo1 + 512 + k);

#pragma unroll
    for (int p = 0; p < 2; ++p) {
      v8f* accp = (p == 0) ? &acc0 : &acc1;
      int nb = n0 + p * 16;
      if (nb < 256) {
#pragma unroll
        for (int r = 0; r < 8; ++r) {
          int m = m0 + r + 8 * half;
          outX[(size_t)(b0 + m) * 256 + nb + lrow] = (*accp)[r];
        }
      } else {
#pragma unroll
        for (int r = 0; r < 8; ++r) {
          int m = m0 + r + 8 * half;
          float v = (*accp)[r];
          float s = 8.0f + 4.8f * (1.0f / (1.0f + __expf(-v)));
          outY[(size_t)(b0 + m) * 128 + (nb - 256) + lrow] = s;
        }
      }
    }
  }
}

// =====================================================================
// Host launcher
// =====================================================================
extern "C" void kernel_launch(void* const* d_in, const int* in_sizes, int n_in,
                              void* d_out, int out_size, void* d_ws, size_t ws_size,
                              hipStream_t stream) {
  (void)in_sizes; (void)n_in; (void)out_size; (void)ws_size;
  const float* u   = (const float*)d_in[0];
  const float* x   = (const float*)d_in[1];
  const float* X   = (const float*)d_in[2];
  const float* Y   = (const float*)d_in[3];
  const float* Z3  = (const float*)d_in[4];
  const float* X3  = (const float*)d_in[5];
  const float* Y3  = (const float*)d_in[6];
  const float* D12 = (const float*)d_in[7];
  const float* B2  = (const float*)d_in[8];
  const float* C2  = (const float*)d_in[9];

  float* ws = (float*)d_ws;
  float* IpM    = ws + 0;        // 96*96
  float* invIpM = ws + 9216;     // 96*96
  float* D22    = ws + 18432;    // 128*96
  float* Rcal   = ws + 30720;    // 96*96
  float* invR   = ws + 39936;    // 96*96
  float* vecr   = ws + 49152;    // 768*96
  float* T1     = ws + 122880;   // 768*96
  float* H      = ws + 196608;   // 768*768
  float* E      = ws + 786432;   // 256*256
  float* invE   = ws + 851968;   // 256*256
  float* Wb     = ws + 917504;   // 256*352
  float* Wout   = ws + 1007616;  // 384*608
  float* D11    = ws + 1241088;  // 256*256
  float* Lam    = ws + 1306624;  // 256
  float* aug    = ws + 1306880;  // up to 256*512 GJ scratch

  auto nb = [](int n) { return (n + 255) / 256; };

  // ---- Phase A: set_param on device ----
  k_build_IpM<<<nb(96 * 96), 256, 0, stream>>>(X3, Y3, Z3, IpM);
  k_gj_inverse<<<1, 1024, 0, stream>>>(IpM, invIpM, aug, 96);
  // D22 top = (I-Mm)inv(I+Mm) = 2*invIpM - IpM@invIpM ; bottom = -2*Z3@invIpM
  k_gemm<<<nb(96 * 96), 256, 0, stream>>>(D22, IpM, invIpM, 96, 96, 96, 96, 96, 96, 0, 0, -1.f, 0.f, 0.f);
  k_scale_add<<<nb(96 * 96), 256, 0, stream>>>(D22, invIpM, 96, 96, 96, 96, 2.f, 1.f);
  k_gemm<<<nb(32 * 96), 256, 0, stream>>>(D22 + 96 * 96, Z3, invIpM, 32, 96, 96, 96, 96, 96, 0, 0, -2.f, 0.f, 0.f);
  // R_cal = I - D22^T D22 ; invR
  k_gemm<<<nb(96 * 96), 256, 0, stream>>>(Rcal, D22, D22, 96, 96, 128, 96, 96, 96, 1, 0, -1.f, 0.f, 1.f);
  k_gj_inverse<<<1, 1024, 0, stream>>>(Rcal, invR, aug, 96);
  // vec_r = [ -C2^T D22 ; -D12 ; B2 ]   (768x96)
  k_gemm<<<nb(256 * 96), 256, 0, stream>>>(vecr, C2, D22, 256, 96, 128, 256, 96, 96, 1, 0, -1.f, 0.f, 0.f);
  k_scale_add<<<nb(256 * 96), 256, 0, stream>>>(vecr + 256 * 96, D12, 96, 96, 256, 96, -1.f, 0.f);
  k_scale_add<<<nb(256 * 96), 256, 0, stream>>>(vecr + 512 * 96, B2, 96, 96, 256, 96, 1.f, 0.f);
  // H = X^T X + eps*I + vecr invR vecr^T + C2^T C2 (top-left)
  k_gemm<<<nb(768 * 96), 256, 0, stream>>>(T1, vecr, invR, 768, 96, 96, 96, 96, 96, 0, 0, 1.f, 0.f, 0.f);
  k_gemm<<<nb(768 * 768), 256, 0, stream>>>(H, X, X, 768, 768, 768, 768, 768, 768, 1, 0, 1.f, 0.f, EPS_C);
  k_gemm<<<nb(768 * 768), 256, 0, stream>>>(H, T1, vecr, 768, 768, 96, 96, 96, 768, 0, 1, 1.f, 1.f, 0.f);
  k_gemm<<<nb(256 * 256), 256, 0, stream>>>(H, C2, C2, 256, 256, 128, 256, 256, 768, 1, 0, 1.f, 1.f, 0.f);
  // E, invE
  k_build_E<<<nb(256 * 256), 256, 0, stream>>>(H, Y, E);
  k_gj_inverse<<<1, 1024, 0, stream>>>(E, invE, aug, 256);
  // Wout rows 0..255 : [invE@H31 | invE@H32 | invE@B2]
  k_gemm<<<nb(256 * 256), 256, 0, stream>>>(Wout + 0, invE, H + 512 * 768 + 0, 256, 256, 256, 256, 768, 608, 0, 0, 1.f, 0.f, 0.f);
  k_gemm<<<nb(256 * 256), 256, 0, stream>>>(Wout + 256, invE, H + 512 * 768 + 256, 256, 256, 256, 256, 768, 608, 0, 0, 1.f, 0.f, 0.f);
  k_gemm<<<nb(256 * 96), 256, 0, stream>>>(Wout + 512, invE, B2, 256, 96, 256, 256, 96, 608, 0, 0, 1.f, 0.f, 0.f);
  // Wout rows 256..383 : [C2 | 0 (D21) | D22]
  k_scale_add<<<nb(128 * 256), 256, 0, stream>>>(Wout + 256 * 608, C2, 608, 256, 128, 256, 1.f, 0.f);
  k_fill<<<nb(128 * 256), 256, 0, stream>>>(Wout + 256 * 608 + 256, 608, 128, 256, 0.f);
  k_scale_add<<<nb(128 * 96), 256, 0, stream>>>(Wout + 256 * 608 + 512, D22, 608, 96, 128, 96, 1.f, 0.f);
  // Wb = [C1 | D12] with C1 = -H21
  k_scale_add<<<nb(256 * 256), 256, 0, stream>>>(Wb, H + 256 * 768, 352, 768, 256, 256, -1.f, 0.f);
  k_scale_add<<<nb(256 * 96), 256, 0, stream>>>(Wb + 256, D12, 352, 96, 256, 96, 1.f, 0.f);
  // D11, Lam
  k_build_D11Lam<<<nb(256 * 256), 256, 0, stream>>>(H, D11, Lam);

  // ---- Phase B: batched WMMA forward ----
  float* outY = (float*)d_out;
  float* outX = outY + (size_t)B_ * 128;
  k_forward<<<B_ / TB, 256, 0, stream>>>(x, u, Wb, Wout, D11, Lam, outY, outX);
}